// ThermalLayer_25984552140899
// MI455X (gfx1250) — compile-verified
//
#include <hip/hip_runtime.h>

typedef float v2f __attribute__((ext_vector_type(2)));
typedef float v8f __attribute__((ext_vector_type(8)));
typedef int   v4i __attribute__((ext_vector_type(4)));

#define K_DIM 512
#define N_DIM 4096
#define HGT 64
#define WID 64
#define CELLS (HGT * WID)

#define AS1 __attribute__((address_space(1)))
#define AS3 __attribute__((address_space(3)))

// ---------------------------------------------------------------------------
// Kernel 1: heat = x @ conductance^T + bias   (exact fp32 via WMMA 16x16x4)
// Block = 256 threads = 8 waves. Each wave: 16 (M) x 64 (N) output strip.
// Block tile: 128 (M) x 64 (N). Grid: (N/64, M/128).
// ---------------------------------------------------------------------------
__global__ __launch_bounds__(256) void thermal_gemm_f32(
    const float* __restrict__ x,      // [M, 512]
    const float* __restrict__ w,      // [4096, 512]
    const float* __restrict__ bias,   // [4096]
    float* __restrict__ out)          // [M, 4096]
{
    const int lane = threadIdx.x & 31;
    const int wave = threadIdx.x >> 5;
    const int row  = lane & 15;   // m (A) / n (B) sub-row within tile
    const int hi   = lane >> 4;   // 0: k,k+1   1: k+2,k+3

    const int m0 = blockIdx.y * 128 + wave * 16;
    const int n0 = blockIdx.x * 64;

    // A fragment source: x[m0+row][k + 2*hi]  (float2)
    const float* aptr = x + (size_t)(m0 + row) * K_DIM + 2 * hi;
    // B fragment source: w[n0 + t*16 + row][k + 2*hi]  (float2), t = 0..3
    const float* bptr = w + (size_t)(n0 + row) * K_DIM + 2 * hi;

    v8f acc[4];
    acc[0] = (v8f){}; acc[1] = (v8f){}; acc[2] = (v8f){}; acc[3] = (v8f){};

#pragma unroll 4
    for (int k = 0; k < K_DIM; k += 4) {
        v2f a  = *(const v2f*)(aptr + k);
        v2f b0 = *(const v2f*)(bptr + k);
        v2f b1 = *(const v2f*)(bptr + 16 * K_DIM + k);
        v2f b2 = *(const v2f*)(bptr + 32 * K_DIM + k);
        v2f b3 = *(const v2f*)(bptr + 48 * K_DIM + k);
        acc[0] = __builtin_amdgcn_wmma_f32_16x16x4_f32(false, a, false, b0,
                                                       (short)0, acc[0], false, false);
        acc[1] = __builtin_amdgcn_wmma_f32_16x16x4_f32(false, a, false, b1,
                                                       (short)0, acc[1], false, false);
        acc[2] = __builtin_amdgcn_wmma_f32_16x16x4_f32(false, a, false, b2,
                                                       (short)0, acc[2], false, false);
        acc[3] = __builtin_amdgcn_wmma_f32_16x16x4_f32(false, a, false, b3,
                                                       (short)0, acc[3], false, false);
    }

    // C/D layout: VGPR v, lanes 0-15 -> (m=v,   n=lane)
    //                     lanes 16-31-> (m=v+8, n=lane-16)
#pragma unroll
    for (int t = 0; t < 4; ++t) {
        const int n = n0 + t * 16 + row;
        const float bv = bias[n];
#pragma unroll
        for (int v = 0; v < 8; ++v) {
            const int m = m0 + v + 8 * hi;
            out[(size_t)m * N_DIM + n] = acc[t][v] + bv;
        }
    }
}

// ---------------------------------------------------------------------------
// Kernel 2: in-place diffusion. One workgroup per 64x64 image, all 10 steps
// in LDS (raw-heat src 16KB + double-buffered field 32KB = 48KB). Boundary
// cells are exactly AMBIENT after every step, so out-of-range neighbors never
// matter for interior cells and boundaries are just written as 25.0.
// Heat map is staged into LDS with GLOBAL_LOAD_ASYNC_TO_LDS_B128 (ASYNCcnt)
// when the builtin is available, bypassing the VGPR round trip.
// ---------------------------------------------------------------------------
__global__ __launch_bounds__(256) void thermal_diffuse(
    float* __restrict__ data,          // [B, 64*64] heat in, temperature out
    const int* __restrict__ steps_ptr)
{
    __shared__ float src[CELLS];       // raw heat (scaled inside the stencil)
    __shared__ float tf[2][CELLS];

    const float r      = 9.7e-5f * 1e-3f / (1e-3f * 1e-3f);   // 0.097
    const float center = 1.0f - 4.0f * r;
    const float sc     = 1e-3f / (2700.0f * 900.0f);          // DT/(RHO*CP)
    const float amb    = 25.0f;

    float* img = data + (size_t)blockIdx.x * CELLS;
    const int steps = steps_ptr[0];

#if __has_builtin(__builtin_amdgcn_global_load_async_to_lds_b128)
    // Async DMA: memory -> LDS, 16B per lane per issue, no VGPR staging.
#pragma unroll
    for (int i = threadIdx.x; i < CELLS / 4; i += 256) {
        __builtin_amdgcn_global_load_async_to_lds_b128(
            (AS1 v4i*)(img + 4 * i), (AS3 v4i*)(src + 4 * i), 0, 0);
    }
#pragma unroll
    for (int i = threadIdx.x; i < CELLS / 4; i += 256) {
        ((float4*)tf[0])[i] = make_float4(0.f, 0.f, 0.f, 0.f);
    }
#if __has_builtin(__builtin_amdgcn_s_wait_asynccnt)
    __builtin_amdgcn_s_wait_asynccnt(0);
#else
    asm volatile("s_wait_asynccnt 0x0" ::: "memory");
#endif
#else
    // Fallback: plain vector loads + DS stores.
#pragma unroll
    for (int i = threadIdx.x; i < CELLS / 4; i += 256) {
        ((float4*)src)[i]   = ((const float4*)img)[i];
        ((float4*)tf[0])[i] = make_float4(0.f, 0.f, 0.f, 0.f);
    }
#endif
    __syncthreads();

    int p = 0;
    for (int s = 0; s < steps; ++s) {
        const float* __restrict__ tc = tf[p];
        float* __restrict__ tn = tf[p ^ 1];
#pragma unroll 4
        for (int idx = threadIdx.x; idx < CELLS; idx += 256) {
            const int i = idx >> 6;
            const int j = idx & 63;
            float v;
            if (i == 0 || i == HGT - 1 || j == 0 || j == WID - 1) {
                v = amb;
            } else {
                v = r * (tc[idx - WID] + tc[idx + WID] + tc[idx - 1] + tc[idx + 1])
                    + center * tc[idx] + sc * src[idx];
            }
            tn[idx] = v;
        }
        __syncthreads();
        p ^= 1;
    }

    // Write result back over the same image (only this WG touches it).
#pragma unroll
    for (int i = threadIdx.x; i < CELLS / 4; i += 256) {
        ((float4*)img)[i] = ((const float4*)tf[p])[i];
    }
}

// ---------------------------------------------------------------------------
extern "C" void kernel_launch(void* const* d_in, const int* in_sizes, int n_in,
                              void* d_out, int out_size, void* d_ws, size_t ws_size,
                              hipStream_t stream)
{
    const float* x     = (const float*)d_in[0];   // [B, 512]
    const float* w     = (const float*)d_in[1];   // [4096, 512]
    const float* bias  = (const float*)d_in[2];   // [4096]
    const int*   steps = (const int*)d_in[3];     // scalar (10)
    float* out = (float*)d_out;                   // [B, 4096]

    const int batch = in_sizes[0] / K_DIM;        // 8192

    dim3 g1(N_DIM / 64, batch / 128);             // (64, 64)
    thermal_gemm_f32<<<g1, 256, 0, stream>>>(x, w, bias, out);

    thermal_diffuse<<<dim3(batch), 256, 0, stream>>>(out, steps);
}